// ChunkedLinearMemoryUpdate_66099546685581
// MI455X (gfx1250) — compile-verified
//
#include <hip/hip_runtime.h>
#include <math.h>

typedef float v2f __attribute__((ext_vector_type(2)));
typedef float v4f __attribute__((ext_vector_type(4)));
typedef float v8f __attribute__((ext_vector_type(8)));

#define D_MODEL 768
#define CHUNK   128

// GEMM1: preds_lin = keys @ W^T  (M=128 tokens, N=768 outs, K=768)
// epilogue: E2T[o][t] = c[t] * (preds_lin - values)   (transposed store)
//           loss partials (sum over this block's 64 columns of err_b^2)
__global__ __launch_bounds__(32)
void k_gemm1(const float* __restrict__ W, const float* __restrict__ bparam,
             const float* __restrict__ keys, const float* __restrict__ values,
             float* __restrict__ E2T, float* __restrict__ lpart) {
  const int lane = threadIdx.x;
  const int lo = lane & 15;
  const int hi = lane >> 4;
  const int M0 = blockIdx.y << 4;   // token tile base (t)
  const int N0 = blockIdx.x << 6;   // 4 output tiles (o), 64 cols

  v8f acc[4] = {};
  const float* arow = keys + (M0 + lo) * D_MODEL + 2 * hi;          // A: keys[t][k]
  const float* brow[4];
  #pragma unroll
  for (int nt = 0; nt < 4; ++nt)
    brow[nt] = W + (N0 + nt * 16 + lo) * D_MODEL + 2 * hi;          // B[k][o] = W[o][k]

  for (int k = 0; k < D_MODEL; k += 4) {
    v2f a = *(const v2f*)(arow + k);
    #pragma unroll
    for (int nt = 0; nt < 4; ++nt) {
      v2f bf = *(const v2f*)(brow[nt] + k);
      acc[nt] = __builtin_amdgcn_wmma_f32_16x16x4_f32(
          false, a, false, bf, (short)0, acc[nt], false, false);
    }
  }

  float lsum[8] = {0.f,0.f,0.f,0.f,0.f,0.f,0.f,0.f};
  #pragma unroll
  for (int nt = 0; nt < 4; ++nt) {
    const int o = N0 + nt * 16 + lo;
    const float bo = bparam[o];
    float e2[8];
    #pragma unroll
    for (int r = 0; r < 8; ++r) {
      const int t = M0 + 8 * hi + r;                   // D layout: VGPR r -> M=r+8*hi
      const float errW = acc[nt][r] - values[t * D_MODEL + o];
      const float errB = errW + bo;
      lsum[r] += errB * errB;
      const float ct = 0.02f * powf(0.891f, (float)(127 - t)); // 2*LR*(0.9*0.99)^(127-t)
      e2[r] = ct * errW;
    }
    float* dst = E2T + o * CHUNK + M0 + 8 * hi;        // contiguous in t -> float4 x2
    *(v4f*)(dst)     = (v4f){e2[0], e2[1], e2[2], e2[3]};
    *(v4f*)(dst + 4) = (v4f){e2[4], e2[5], e2[6], e2[7]};
  }

  // reduce loss partials across the 16 lanes of each hi-group, write per-block slot
  #pragma unroll
  for (int r = 0; r < 8; ++r) {
    float s = lsum[r];
    for (int m = 1; m < 16; m <<= 1) s += __shfl_xor(s, m, 32);
    if (lo == 0) lpart[(M0 + 8 * hi + r) * 12 + blockIdx.x] = s;
  }
}

// GEMM2: G[o][i] = sum_t E2T[o][t] * keys[t][i]   (M=768, N=768, K=128)
// epilogue: mW_new = 0.9^128*mW - G ; W_new = 0.99^128*W + mW_new
__global__ __launch_bounds__(32)
void k_gemm2(const float* __restrict__ E2T, const float* __restrict__ keys,
             const float* __restrict__ W, const float* __restrict__ mW,
             float* __restrict__ W_new, float* __restrict__ mW_new) {
  const int lane = threadIdx.x;
  const int lo = lane & 15;
  const int hi = lane >> 4;
  const int M0 = blockIdx.y << 4;   // o tile
  const int N0 = blockIdx.x << 6;   // i tiles (64 cols)

  v8f acc[4] = {};
  const float* arow = E2T + (M0 + lo) * CHUNK + 2 * hi;   // A[o][t], contiguous
  for (int k = 0; k < CHUNK; k += 4) {
    v2f a = *(const v2f*)(arow + k);
    const float* k0 = keys + (k + 2 * hi) * D_MODEL;      // B[t][i] = keys rows
    const float* k1 = keys + (k + 2 * hi + 1) * D_MODEL;
    #pragma unroll
    for (int nt = 0; nt < 4; ++nt) {
      const int n = N0 + nt * 16 + lo;
      v2f bf = (v2f){k0[n], k1[n]};
      acc[nt] = __builtin_amdgcn_wmma_f32_16x16x4_f32(
          false, a, false, bf, (short)0, acc[nt], false, false);
    }
  }

  const float mdec = powf(0.9f, 128.0f);
  const float btot = powf(0.99f, 128.0f);
  #pragma unroll
  for (int nt = 0; nt < 4; ++nt) {
    #pragma unroll
    for (int r = 0; r < 8; ++r) {
      const int o = M0 + 8 * hi + r;
      const int i = N0 + nt * 16 + lo;
      const float g = acc[nt][r];
      const float nm = mdec * mW[o * D_MODEL + i] - g;
      mW_new[o * D_MODEL + i] = nm;
      W_new[o * D_MODEL + i]  = btot * W[o * D_MODEL + i] + nm;
    }
  }
}

// bias update: sum_t c[t]*err_b[t][o] = colsum(E2T[o][:]) + b[o]*sum_t c[t]
__global__ void k_bias(const float* __restrict__ E2T, const float* __restrict__ bparam,
                       const float* __restrict__ mb,
                       float* __restrict__ b_new, float* __restrict__ mb_new) {
  const int o = blockIdx.x * blockDim.x + threadIdx.x;
  if (o >= D_MODEL) return;
  const float* col = E2T + o * CHUNK;
  float s = 0.f;
  #pragma unroll 4
  for (int t = 0; t < CHUNK; ++t) s += col[t];
  const float csum = 0.02f * (1.0f - powf(0.891f, 128.0f)) / (1.0f - 0.891f);
  const float mdec = powf(0.9f, 128.0f);
  const float btot = powf(0.99f, 128.0f);
  const float nm = mdec * mb[o] - (s + bparam[o] * csum);
  mb_new[o] = nm;
  b_new[o]  = btot * bparam[o] + nm;
}

__global__ void k_loss(const float* __restrict__ lpart, float* __restrict__ losses) {
  const int t = threadIdx.x;
  if (t < CHUNK) {
    float s = 0.f;
    #pragma unroll
    for (int j = 0; j < 12; ++j) s += lpart[t * 12 + j];
    losses[t] = s * (1.0f / (float)D_MODEL);
  }
}

extern "C" void kernel_launch(void* const* d_in, const int* in_sizes, int n_in,
                              void* d_out, int out_size, void* d_ws, size_t ws_size,
                              hipStream_t stream) {
  (void)in_sizes; (void)n_in; (void)out_size; (void)ws_size;
  const float* W      = (const float*)d_in[0];
  const float* bparam = (const float*)d_in[1];
  const float* keys   = (const float*)d_in[2];
  const float* values = (const float*)d_in[3];
  const float* mW     = (const float*)d_in[4];
  const float* mb     = (const float*)d_in[5];

  float* out     = (float*)d_out;
  float* W_new   = out;                              // 768*768
  float* b_new   = W_new  + D_MODEL * D_MODEL;       // 768
  float* mW_new  = b_new  + D_MODEL;                 // 768*768
  float* mb_new  = mW_new + D_MODEL * D_MODEL;       // 768
  float* losses  = mb_new + D_MODEL;                 // 128

  float* E2T   = (float*)d_ws;                       // 768*128 floats
  float* lpart = E2T + D_MODEL * CHUNK;              // 128*12 floats

  k_gemm1<<<dim3(12, 8),  32, 0, stream>>>(W, bparam, keys, values, E2T, lpart);
  k_gemm2<<<dim3(12, 48), 32, 0, stream>>>(E2T, keys, W, mW, W_new, mW_new);
  k_bias <<<3, 256, 0, stream>>>(E2T, bparam, mb, b_new, mb_new);
  k_loss <<<1, 128, 0, stream>>>(lpart, losses);
}